// TAHIN_52458730553640
// MI455X (gfx1250) — compile-verified
//
#include <hip/hip_runtime.h>
#include <hip/hip_bf16.h>
#include <stdint.h>

// LightGCN propagation for MI455X (gfx1250, wave32).
// SpMM is L2-resident (77 MB working set << 192 MB L2); bound by L2 bandwidth
// and f32 atomic throughput, not FLOPs -> no WMMA applicability (no dense
// GEMM exists in the reference). CDNA5 paths used:
// global_load_async_to_lds_b32 + s_wait_asynccnt (ASYNCcnt DMA edge staging)
// and native no-return global_atomic_add_f32 (STOREcnt-tracked).

#define EMB   64
#define TPB   256
#define CHUNK 256   // edges staged per block

// ---------------- scatter (the hot kernel, placed first for disasm) ----------------
// For each edge e: msg[rows[e], :] += dis[rows[e]]*dis[cols[e]] * x[cols[e], :]
// One block = one chunk of 256 edges. Indices staged into LDS with CDNA5
// async global->LDS DMA; g computed once per edge; then one wave (32 lanes x
// float2) per edge slot does coalesced 256B gathers of x rows (L2 hits) and
// f32 atomic adds into msg (L2 atomics).
__global__ void scatter_kernel(const int* __restrict__ rows,
                               const int* __restrict__ cols,
                               const float* __restrict__ dis,
                               const float* __restrict__ x,
                               float* __restrict__ msg, int E) {
    __shared__ int   s_row[CHUNK];
    __shared__ int   s_col[CHUNK];
    __shared__ float s_g[CHUNK];

    const int tid  = threadIdx.x;
    const long long base = (long long)blockIdx.x * CHUNK;
    const long long rem  = (long long)E - base;
    const int nval = (rem < CHUNK) ? (int)rem : CHUNK;

    if (tid < nval) {
        // LDS byte offsets (low 32 bits of the flat shared address)
        unsigned lds_r = (unsigned)(uintptr_t)&s_row[tid];
        unsigned lds_c = (unsigned)(uintptr_t)&s_col[tid];
        const int* gr = rows + base + tid;
        const int* gc = cols + base + tid;
        // CDNA5 async DMA: global -> LDS, tracked by ASYNCcnt
        asm volatile("global_load_async_to_lds_b32 %0, %1, off"
                     :: "v"(lds_r), "v"(gr) : "memory");
        asm volatile("global_load_async_to_lds_b32 %0, %1, off"
                     :: "v"(lds_c), "v"(gc) : "memory");
    }
    asm volatile("s_wait_asynccnt 0x0" ::: "memory");
    __syncthreads();

    if (tid < nval) {
        s_g[tid] = dis[s_row[tid]] * dis[s_col[tid]];
    }
    __syncthreads();

    const int lane = tid & 31;          // 32 lanes/edge, float2 per lane
    const int sub  = tid >> 5;          // 8 concurrent edge slots per block
    const float2* __restrict__ x2 = (const float2*)x;
    for (int e = sub; e < nval; e += TPB / 32) {
        const int   r = s_row[e];
        const int   c = s_col[e];
        const float g = s_g[e];
        const float2 xv = x2[(size_t)c * (EMB / 2) + lane];
        float* dst = &msg[(size_t)r * EMB + lane * 2];
        __hip_atomic_fetch_add(dst + 0, g * xv.x,
                               __ATOMIC_RELAXED, __HIP_MEMORY_SCOPE_AGENT);
        __hip_atomic_fetch_add(dst + 1, g * xv.y,
                               __ATOMIC_RELAXED, __HIP_MEMORY_SCOPE_AGENT);
    }
}

// ---------------- elementwise helpers ----------------

__global__ void zero_scalar_kernel(float* __restrict__ p, long long n) {
    long long i = (long long)blockIdx.x * blockDim.x + threadIdx.x;
    long long stride = (long long)gridDim.x * blockDim.x;
    for (; i < n; i += stride) p[i] = 0.0f;
}

__global__ void zero_f4_kernel(float4* __restrict__ p, long long n4) {
    long long i = (long long)blockIdx.x * blockDim.x + threadIdx.x;
    long long stride = (long long)gridDim.x * blockDim.x;
    float4 z; z.x = z.y = z.z = z.w = 0.0f;
    for (; i < n4; i += stride) p[i] = z;
}

// x = concat(user_emb, item_emb); acc = x
__global__ void init_x_acc_kernel(const float4* __restrict__ ue,
                                  const float4* __restrict__ ie,
                                  float4* __restrict__ x,
                                  float4* __restrict__ acc,
                                  long long nu4, long long total4) {
    long long i = (long long)blockIdx.x * blockDim.x + threadIdx.x;
    long long stride = (long long)gridDim.x * blockDim.x;
    for (; i < total4; i += stride) {
        float4 v = (i < nu4) ? ue[i] : ie[i - nu4];
        x[i]   = v;
        acc[i] = v;
    }
}

// deg[rows[e]] += 1.0f  (native global_atomic_add_f32, agent scope, L2-resident)
__global__ void deg_kernel(const int* __restrict__ rows,
                           float* __restrict__ deg, int E) {
    int i = blockIdx.x * blockDim.x + threadIdx.x;
    if (i < E) {
        __hip_atomic_fetch_add(&deg[rows[i]], 1.0f,
                               __ATOMIC_RELAXED, __HIP_MEMORY_SCOPE_AGENT);
    }
}

__global__ void rsqrt_kernel(const float* __restrict__ deg,
                             float* __restrict__ dis, int n) {
    int i = blockIdx.x * blockDim.x + threadIdx.x;
    if (i < n) {
        float d = deg[i];
        dis[i] = (d > 0.0f) ? rsqrtf(d) : 0.0f;
    }
}

// x += msg; acc += x   (float4, streams through HBM/L2)
__global__ void combine_kernel(float4* __restrict__ x,
                               const float4* __restrict__ msg,
                               float4* __restrict__ acc,
                               long long total4) {
    long long i = (long long)blockIdx.x * blockDim.x + threadIdx.x;
    long long stride = (long long)gridDim.x * blockDim.x;
    for (; i < total4; i += stride) {
        float4 xv = x[i];
        float4 m  = msg[i];
        float4 a  = acc[i];
        xv.x += m.x; xv.y += m.y; xv.z += m.z; xv.w += m.w;
        a.x += xv.x; a.y += xv.y; a.z += xv.z; a.w += xv.w;
        x[i]   = xv;
        acc[i] = a;
    }
}

// ---------------- host launcher ----------------

extern "C" void kernel_launch(void* const* d_in, const int* in_sizes, int n_in,
                              void* d_out, int out_size, void* d_ws, size_t ws_size,
                              hipStream_t stream) {
    const float* ue   = (const float*)d_in[0];   // user_emb  (n_users*64)
    const float* ie   = (const float*)d_in[1];   // item_emb  (n_items*64)
    const int*   rows = (const int*)d_in[2];     // 2*N_BASE_EDGES
    const int*   cols = (const int*)d_in[3];
    float*       acc  = (float*)d_out;           // acc, flat (n*64)

    const int n_users = in_sizes[0] / EMB;
    const int n_items = in_sizes[1] / EMB;
    const int E       = in_sizes[2];
    const int n       = n_users + n_items;
    const long long total  = (long long)n * EMB;
    const long long total4 = total / 4;          // EMB=64 -> divisible by 4
    const long long nu4    = (long long)n_users * EMB / 4;

    // workspace layout (floats): deg[n] | dis[n] | x[n*64] | msg[n*64]
    float* deg = (float*)d_ws;
    float* dis = deg + n;
    float* x   = dis + n;
    float* msg = x + total;

    const int gb_elem   = (int)((total4 + TPB - 1) / TPB);
    const int gb_n      = (n + TPB - 1) / TPB;
    const int gb_E      = (E + TPB - 1) / TPB;
    const int gb_chunks = (E + CHUNK - 1) / CHUNK;

    zero_scalar_kernel<<<gb_n, TPB, 0, stream>>>(deg, n);
    init_x_acc_kernel<<<gb_elem, TPB, 0, stream>>>(
        (const float4*)ue, (const float4*)ie, (float4*)x, (float4*)acc,
        nu4, total4);
    deg_kernel<<<gb_E, TPB, 0, stream>>>(rows, deg, E);
    rsqrt_kernel<<<gb_n, TPB, 0, stream>>>(deg, dis, n);

    for (int layer = 0; layer < 2; ++layer) {   // N_LAYERS = 2
        zero_f4_kernel<<<gb_elem, TPB, 0, stream>>>((float4*)msg, total4);
        scatter_kernel<<<gb_chunks, TPB, 0, stream>>>(rows, cols, dis, x, msg, E);
        combine_kernel<<<gb_elem, TPB, 0, stream>>>(
            (float4*)x, (const float4*)msg, (float4*)acc, total4);
    }
}